// UnfiltBackproj3DForLoop_6416681140696
// MI455X (gfx1250) — compile-verified
//
#include <hip/hip_runtime.h>

typedef __attribute__((ext_vector_type(2))) float v2f;
typedef __attribute__((ext_vector_type(8))) float v8f;

#define B_DIM 8
#define K_DIM 128
#define N2 16
#define N3 16
#define P_DIM 1024
#define QR 16                     // q*4 + r
#define LDS_STRIDE (K_DIM + 2)    // pad 2 floats/row to rotate LDS banks

// out[n, p, yy*4+q, z*4+r]  (flattened (8,1,1024,64,64))
// GEMM: D[z, c] = sum_k y[n,k,yy,z] * W[k, p, c]   (c = q*4+r)
__global__ __launch_bounds__(256)
void backproj_wmma_f32_kernel(const float* __restrict__ y,
                              const float* __restrict__ W,
                              float* __restrict__ out) {
    __shared__ float Ash[N3 * LDS_STRIDE];   // [z][k], 16 x 130 f32

    const int tid    = threadIdx.x;
    const int waveId = tid >> 5;
    const int lane   = tid & 31;
    const int lane16 = lane & 15;
    const int half   = lane >> 4;

    const int rowTile = blockIdx.y;   // 0..127 : (n, yy)
    const int n  = rowTile >> 4;      // 0..7
    const int yy = rowTile & 15;      // 0..15

    // ---- stage A tile (16 z-rows x 128 k) into LDS ----
    // y[n,k,0,yy,z] at n*(128*256) + k*256 + yy*16 + z
    const float* ybase = y + (size_t)n * (K_DIM * N2 * N3) + yy * N3;
#pragma unroll
    for (int i = 0; i < 8; ++i) {
        int f = i * 256 + tid;        // 0..2047
        int z = f & 15;
        int k = f >> 4;
        Ash[z * LDS_STRIDE + k] = ybase[(size_t)k * (N2 * N3) + z];
    }
    __syncthreads();

    // ---- two p-tiles per wave: block covers 16 consecutive p ----
    const int p0 = blockIdx.x * 16 + waveId * 2;

    // B element (k, c=lane16) lives at W[k*16384 + p*16 + c]
    const float* Wl0 = W + (size_t)p0 * QR + lane16;
    const float* Wl1 = Wl0 + QR;

    v8f c0 = {};
    v8f c1 = {};

    // A fragment source: row z = lane16, k = k0 + 2*half + {0,1} (contiguous -> b64)
    const float* arow = &Ash[lane16 * LDS_STRIDE + 2 * half];

#pragma unroll
    for (int kk = 0; kk < 32; ++kk) {
        const int k0 = kk * 4;
        v2f a = *(const v2f*)(arow + k0);               // ds_load_b64

        const size_t koff = (size_t)(k0 + 2 * half) * (P_DIM * QR);
        v2f b0, b1;
        b0.x = Wl0[koff];
        b0.y = Wl0[koff + (size_t)(P_DIM * QR)];
        b1.x = Wl1[koff];
        b1.y = Wl1[koff + (size_t)(P_DIM * QR)];

        // D = A(16x4) * B(4x16) + C, full fp32 (matches reference arithmetic)
        c0 = __builtin_amdgcn_wmma_f32_16x16x4_f32(false, a, false, b0,
                                                   (short)0, c0, false, false);
        c1 = __builtin_amdgcn_wmma_f32_16x16x4_f32(false, a, false, b1,
                                                   (short)0, c1, false, false);
    }

    // ---- store: D element (row z = v + 8*half, col c = lane16) ----
    // addr = ((n*1024 + p)*64 + yy*4 + (c>>2))*64 + z*4 + (c&3)
    const int q = lane16 >> 2;
    const int r = lane16 & 3;
    float* ob0 = out + (((size_t)(n * P_DIM + p0) * 64 + yy * 4 + q) * 64) + r;
    float* ob1 = ob0 + 64 * 64;
#pragma unroll
    for (int v = 0; v < 8; ++v) {
        int z = v + 8 * half;
        ob0[(size_t)z * 4] = c0[v];
        ob1[(size_t)z * 4] = c1[v];
    }
}

extern "C" void kernel_launch(void* const* d_in, const int* in_sizes, int n_in,
                              void* d_out, int out_size, void* d_ws, size_t ws_size,
                              hipStream_t stream) {
    const float* y = (const float*)d_in[0];   // (8,128,1,16,16) f32
    const float* W = (const float*)d_in[1];   // (128,1,1024,4,4) f32
    float* out = (float*)d_out;               // (8,1,1024,64,64) f32

    dim3 grid(P_DIM / 16, B_DIM * N2);        // (64, 128) blocks
    backproj_wmma_f32_kernel<<<grid, 256, 0, stream>>>(y, W, out);
}